// LogAdd_8804682957467
// MI455X (gfx1250) — compile-verified
//
#include <hip/hip_runtime.h>
#include <stdint.h>

#define B_ 4
#define T_ 256
#define U_ 128
#define V_ 1024

typedef __attribute__((ext_vector_type(4))) float v4f;

__device__ __forceinline__ float fast_exp2(float a) {
#if __has_builtin(__builtin_amdgcn_exp2f)
  return __builtin_amdgcn_exp2f(a);   // v_exp_f32
#else
  return exp2f(a);
#endif
}

__device__ __forceinline__ float fast_log2(float a) {
#if __has_builtin(__builtin_amdgcn_logf)
  return __builtin_amdgcn_logf(a);    // v_log_f32 (log base 2)
#else
  return log2f(a);
#endif
}

// Per-lane async copy of 16B from global to LDS (CDNA5 async-tensor path).
__device__ __forceinline__ void async_copy_b128(uint32_t lds_off, const void* gaddr) {
  asm volatile("global_load_async_to_lds_b128 %0, %1, off"
               :: "v"(lds_off), "v"(gaddr)
               : "memory");
}

__global__ __launch_bounds__(256) void rnnt_log_softmax_kernel(
    const float* __restrict__ f, const float* __restrict__ g,
    float* __restrict__ out)
{
  // 8 waves * 2 buffers * 1024 floats = 64 KB LDS
  __shared__ __align__(16) float gbuf[8 * 2 * V_];

  const int tid  = threadIdx.x;
  const int lane = tid & 31;
  const int wave = tid >> 5;

  const int bt = blockIdx.x;          // 0 .. B*T-1
  const int b  = bt / T_;

  // ---- load f row (b,t) into registers: chunk k covers v = k*128 + lane*4 ..+3
  const float* frow = f + (size_t)bt * V_;
  v4f fx[8];
#pragma unroll
  for (int k = 0; k < 8; ++k)
    fx[k] = *(const v4f*)(frow + k * 128 + lane * 4);

  const float* gB = g + (size_t)b * U_ * V_;
  float* outB = out + (size_t)bt * U_ * V_;

  // LDS byte offset of this wave's double buffer (low 32 bits of generic LDS ptr)
  const uint32_t lds_wave = (uint32_t)(uintptr_t)(&gbuf[wave * 2 * V_]);

  // ---- prefetch first g row (u = wave) into buffer 0
  {
    const char* src = (const char*)(gB + (size_t)wave * V_) + lane * 16;
    const uint32_t dst = lds_wave + (uint32_t)lane * 16u;
#pragma unroll
    for (int k = 0; k < 8; ++k)
      async_copy_b128(dst + (uint32_t)k * 512u, src + k * 512);
  }

  const float L2E = 1.4426950408889634f;   // log2(e)
  const float LN2 = 0.6931471805599453f;   // ln(2)

  for (int i = 0; i < 16; ++i) {
    const int u = wave + i * 8;
    const uint32_t cur = (uint32_t)(i & 1);

    if (i + 1 < 16) {
      // prefetch next row into the other buffer, then wait for current row
      const char* src = (const char*)(gB + (size_t)(u + 8) * V_) + lane * 16;
      const uint32_t dst = lds_wave + (1u - cur) * (V_ * 4u) + (uint32_t)lane * 16u;
#pragma unroll
      for (int k = 0; k < 8; ++k)
        async_copy_b128(dst + (uint32_t)k * 512u, src + k * 512);
      asm volatile("s_wait_asynccnt 8" ::: "memory");   // first 8 (current row) done
    } else {
      asm volatile("s_wait_asynccnt 0" ::: "memory");
    }

    // ---- x = f + g (from LDS), row max
    const float* gl = &gbuf[wave * 2 * V_ + cur * V_];
    v4f x[8];
    float m = -3.402823466e38f;
#pragma unroll
    for (int k = 0; k < 8; ++k) {
      v4f gv = *(const v4f*)(gl + k * 128 + lane * 4);   // ds_load_b128
      v4f xv = fx[k] + gv;
      x[k] = xv;
      m = fmaxf(m, fmaxf(fmaxf(xv[0], xv[1]), fmaxf(xv[2], xv[3])));
    }
    // wave32 max reduction (no barriers: one wave owns the whole vocab row)
#pragma unroll
    for (int off = 16; off > 0; off >>= 1)
      m = fmaxf(m, __shfl_xor(m, off, 32));

    // ---- sum of exp(x - m), base-2 hardware transcendentals
    float s = 0.0f;
#pragma unroll
    for (int k = 0; k < 8; ++k) {
      s += fast_exp2((x[k][0] - m) * L2E);
      s += fast_exp2((x[k][1] - m) * L2E);
      s += fast_exp2((x[k][2] - m) * L2E);
      s += fast_exp2((x[k][3] - m) * L2E);
    }
#pragma unroll
    for (int off = 16; off > 0; off >>= 1)
      s += __shfl_xor(s, off, 32);

    const float lse = m + LN2 * fast_log2(s);   // log-sum-exp

    // ---- streaming non-temporal b128 stores (don't pollute L2; output >> L2)
    float* orow = outB + (size_t)u * V_;
#pragma unroll
    for (int k = 0; k < 8; ++k) {
      v4f o = x[k] - lse;
      __builtin_nontemporal_store(o, (v4f*)(orow + k * 128 + lane * 4));
    }
  }
}

extern "C" void kernel_launch(void* const* d_in, const int* in_sizes, int n_in,
                              void* d_out, int out_size, void* d_ws, size_t ws_size,
                              hipStream_t stream) {
  const float* f = (const float*)d_in[0];   // [B,T,V]
  const float* g = (const float*)d_in[1];   // [B,U,V]
  float* out = (float*)d_out;               // [B,T,U,V]
  (void)in_sizes; (void)n_in; (void)out_size; (void)d_ws; (void)ws_size;

  rnnt_log_softmax_kernel<<<dim3(B_ * T_), dim3(256), 0, stream>>>(f, g, out);
}